// GACsol_18794776887363
// MI455X (gfx1250) — compile-verified
//
#include <hip/hip_runtime.h>

typedef __attribute__((ext_vector_type(2))) float v2f;
typedef __attribute__((ext_vector_type(8))) float v8f;

#define N_NODES 16384
#define N_EDGES 131072
#define F_IN    128
#define HEADS   16
#define HD      1024
#define K_POOL  8192

#if defined(__gfx1250__) && __has_builtin(__builtin_amdgcn_global_load_async_to_lds_b128)
#define HAS_ASYNC_LDS 1
#define AS1 __attribute__((address_space(1)))
#define AS3 __attribute__((address_space(3)))
typedef int v4i_b __attribute__((vector_size(4 * sizeof(int))));
#endif

__device__ __forceinline__ void wait_async_all() {
#if defined(HAS_ASYNC_LDS)
#if __has_builtin(__builtin_amdgcn_s_wait_asynccnt)
  __builtin_amdgcn_s_wait_asynccnt(0);
#else
  asm volatile("s_wait_asynccnt 0" ::: "memory");
#endif
#endif
}

__device__ __forceinline__ void atomicMaxF(float* addr, float val) {
  // init value must be -inf; classic monotonic-bits trick
  if (val >= 0.0f) atomicMax((int*)addr, __float_as_int(val));
  else             atomicMin((unsigned int*)addr, __float_as_uint(val));
}

// ---------------- init: -inf maxes, zero sums ----------------
__global__ void __launch_bounds__(256) init_k(float* lmax, float* denom,
                                              float* smax, float* wsum, float* numer) {
  size_t i = (size_t)blockIdx.x * 256 + threadIdx.x;
  if (i < (size_t)N_NODES * HD) {
    smax[i]  = -__builtin_inff();
    wsum[i]  = 0.0f;
    numer[i] = 0.0f;
  }
  if (i < (size_t)N_NODES * HEADS) {
    lmax[i]  = -__builtin_inff();
    denom[i] = 0.0f;
  }
}

// ---------------- xl = x@Wl, xr = x@Wr via V_WMMA_F32_16X16X4_F32 ----------------
// grid: (1024 M-tiles * 8 N-groups, 2), block 256 (8 waves). Wave w -> N tile tng*8+w.
// A panel (16x128 fp32, 8KB) and B panel (128x128 fp32, 64KB) staged in LDS,
// via GLOBAL_LOAD_ASYNC_TO_LDS_B128 when the toolchain exposes it.
__global__ void __launch_bounds__(256)
gemm_xw_k(const float* __restrict__ x, const float* __restrict__ Wl,
          const float* __restrict__ Wr, float* __restrict__ xl, float* __restrict__ xr) {
  __shared__ float As[16 * F_IN];    // 8 KB
  __shared__ float Bs[128 * 128];    // 64 KB
  const int lin = blockIdx.x;
  const int tm  = lin >> 3;                      // 0..1023
  const int tng = lin & 7;                       // N group of 8 tiles (128 cols)
  const float* W   = (blockIdx.y == 0) ? Wl : Wr;
  float*       out = (blockIdx.y == 0) ? xl : xr;
  const int t = threadIdx.x;

  const float* gA = x + (size_t)tm * 16 * F_IN;          // 16x128 tile, contiguous
  const float* gB = W + tng * 128;                        // rows k: stride HD

#if defined(HAS_ASYNC_LDS)
  {
    // A: 2048 floats = 512 b128 transfers; 2 per thread
#pragma unroll
    for (int j = 0; j < 2; ++j) {
      int i = j * 256 + t;
      __builtin_amdgcn_global_load_async_to_lds_b128(
          (AS1 v4i_b*)(gA + i * 4), (AS3 v4i_b*)(As + i * 4), 0, 0);
    }
    // B: 128x128 floats = 4096 b128 transfers; 16 per thread (coalesced along n)
#pragma unroll
    for (int j = 0; j < 16; ++j) {
      int i = j * 256 + t;                    // b128 id
      int row = i >> 5, c4 = (i & 31) * 4;
      __builtin_amdgcn_global_load_async_to_lds_b128(
          (AS1 v4i_b*)(gB + (size_t)row * HD + c4),
          (AS3 v4i_b*)(Bs + row * 128 + c4), 0, 0);
    }
  }
  wait_async_all();
#else
  {
    int row  = t >> 4;
    int col0 = (t & 15) * 8;
    const float* srcp = gA + row * F_IN + col0;
    float* dstp = As + row * F_IN + col0;
#pragma unroll
    for (int j = 0; j < 8; ++j) dstp[j] = srcp[j];
#pragma unroll
    for (int j = 0; j < 16; ++j) {
      int i = j * 256 + t;
      int brow = i >> 5, c4 = (i & 31) * 4;
      *(float4*)(Bs + brow * 128 + c4) =
          *(const float4*)(gB + (size_t)brow * HD + c4);
    }
  }
#endif
  __syncthreads();

  const int wave = t >> 5;
  const int lane = t & 31;
  const int tn   = tng * 8 + wave;               // 0..63
  const int mloc = lane & 15;                    // A: M index; B/C: N index
  const int koff = (lane < 16) ? 0 : 2;          // lane-half K offset (A and B)
  const int col  = tn * 16 + mloc;

  v8f c = {};
  const float* arow = As + mloc * F_IN + koff;
  const float* brow = Bs + koff * 128 + wave * 16 + mloc;
#pragma unroll 4
  for (int k = 0; k < F_IN / 4; ++k) {
    v2f a, b;
    a.x = arow[k * 4 + 0];
    a.y = arow[k * 4 + 1];
    b.x = brow[(k * 4 + 0) * 128];
    b.y = brow[(k * 4 + 1) * 128];
    c = __builtin_amdgcn_wmma_f32_16x16x4_f32(false, a, false, b, (short)0, c,
                                              false, false);
  }
  const int rbase = tm * 16 + ((lane < 16) ? 0 : 8);
#pragma unroll
  for (int r = 0; r < 8; ++r)
    out[(size_t)(rbase + r) * HD + col] = c[r];
}

// ---------------- per-edge attention logits ----------------
__global__ void __launch_bounds__(256)
edge_logits_k(const float* __restrict__ xl, const float* __restrict__ xr,
              const int* __restrict__ ei, const float* __restrict__ att,
              float* __restrict__ logits) {
  __shared__ float red[256];
  const int e   = blockIdx.x;
  const int t   = threadIdx.x;
  const int src = ei[e];
  const int dst = ei[N_EDGES + e];
  const int f0  = t * 4;
  const float* pl = xl + (size_t)src * HD + f0;
  const float* pr = xr + (size_t)dst * HD + f0;
  float s = 0.0f;
#pragma unroll
  for (int j = 0; j < 4; ++j) {
    float v = pl[j] + pr[j];
    v = (v > 0.0f) ? v : 0.2f * v;               // leaky_relu(0.2)
    s += v * att[f0 + j];
  }
  red[t] = s;
  __syncthreads();
  if (t < HEADS) {
    float acc = 0.0f;
#pragma unroll
    for (int j = 0; j < 16; ++j) acc += red[t * 16 + j];
    logits[(size_t)e * HEADS + t] = acc;
  }
}

// ---------------- segment max of logits over dst ----------------
__global__ void __launch_bounds__(256)
edge_lmax_k(const float* __restrict__ logits, const int* __restrict__ ei,
            float* __restrict__ lmax) {
  int i = blockIdx.x * 256 + threadIdx.x;
  if (i >= N_EDGES * HEADS) return;
  int e = i >> 4, hh = i & 15;
  int dst = ei[N_EDGES + e];
  atomicMaxF(&lmax[(size_t)dst * HEADS + hh], logits[i]);
}

// ---------------- ex = exp(logit - lmax), denom = segment_sum ----------------
__global__ void __launch_bounds__(256)
edge_exp_k(float* __restrict__ logits, const int* __restrict__ ei,
           const float* __restrict__ lmax, float* __restrict__ denom) {
  int i = blockIdx.x * 256 + threadIdx.x;
  if (i >= N_EDGES * HEADS) return;
  int e = i >> 4, hh = i & 15;
  int dst = ei[N_EDGES + e];
  float ex = __expf(logits[i] - lmax[(size_t)dst * HEADS + hh]);
  logits[i] = ex;                                 // reuse buffer as ex
  atomicAdd(&denom[(size_t)dst * HEADS + hh], ex);
}

// ---------------- SoftmaxAggregation pass 1: smax = segmax(t*m) ----------------
__global__ void __launch_bounds__(256)
pass_smax_k(const float* __restrict__ xl, const int* __restrict__ ei,
            const float* __restrict__ ex, const float* __restrict__ denom,
            const float* __restrict__ tptr, float* __restrict__ smax) {
  const int e = blockIdx.x, t = threadIdx.x;
  const int src = ei[e], dst = ei[N_EDGES + e];
  const int f0 = t * 4, hh = t >> 4;
  const float alpha = ex[(size_t)e * HEADS + hh] / denom[(size_t)dst * HEADS + hh];
  const float tt = tptr[0];
  const float* pl = xl + (size_t)src * HD + f0;
  float* ps = smax + (size_t)dst * HD + f0;
#pragma unroll
  for (int j = 0; j < 4; ++j) atomicMaxF(&ps[j], tt * (alpha * pl[j]));
}

// ---------------- pass 2: wsum += exp(t*m - smax); numer += w*m ----------------
__global__ void __launch_bounds__(256)
pass_accum_k(const float* __restrict__ xl, const int* __restrict__ ei,
             const float* __restrict__ ex, const float* __restrict__ denom,
             const float* __restrict__ tptr, const float* __restrict__ smax,
             float* __restrict__ wsum, float* __restrict__ numer) {
  const int e = blockIdx.x, t = threadIdx.x;
  const int src = ei[e], dst = ei[N_EDGES + e];
  const int f0 = t * 4, hh = t >> 4;
  const float alpha = ex[(size_t)e * HEADS + hh] / denom[(size_t)dst * HEADS + hh];
  const float tt = tptr[0];
  const float* pl = xl + (size_t)src * HD + f0;
  const float* ps = smax + (size_t)dst * HD + f0;
  float* pw = wsum  + (size_t)dst * HD + f0;
  float* pn = numer + (size_t)dst * HD + f0;
#pragma unroll
  for (int j = 0; j < 4; ++j) {
    float m = alpha * pl[j];
    float w = __expf(tt * m - ps[j]);
    atomicAdd(&pw[j], w);
    atomicAdd(&pn[j], w * m);
  }
}

// ---------------- h = relu(numer/wsum + b1); zero numer for reuse as nbr -------
__global__ void __launch_bounds__(256)
node_h_k(float* __restrict__ numer, const float* __restrict__ wsum,
         const float* __restrict__ b1, float* __restrict__ hbuf) {
  size_t i = (size_t)blockIdx.x * 256 + threadIdx.x;
  if (i >= (size_t)N_NODES * HD) return;
  float ws = wsum[i];
  float a = (ws > 0.0f) ? (numer[i] / ws) : 0.0f;
  a += b1[i & (HD - 1)];
  hbuf[i] = (a > 0.0f) ? a : 0.0f;
  numer[i] = 0.0f;                                // becomes nbr accumulator
}

// ---------------- nbr = segment_sum(h[src]) over dst ----------------
__global__ void __launch_bounds__(256)
pass_nbr_k(const float* __restrict__ hbuf, const int* __restrict__ ei,
           float* __restrict__ nbr) {
  const int e = blockIdx.x, t = threadIdx.x;
  const int src = ei[e], dst = ei[N_EDGES + e];
  const int f0 = t * 4;
  const float* ph = hbuf + (size_t)src * HD + f0;
  float* pn = nbr + (size_t)dst * HD + f0;
#pragma unroll
  for (int j = 0; j < 4; ++j) atomicAdd(&pn[j], ph[j]);
}

// ---------------- score = tanh(nbr@w_rel + h@w_root + b_sag) ----------------
__global__ void __launch_bounds__(256)
node_score_k(const float* __restrict__ nbr, const float* __restrict__ hbuf,
             const float* __restrict__ w_rel, const float* __restrict__ w_root,
             const float* __restrict__ b_sag, float* __restrict__ score) {
  __shared__ float red[256];
  const int n = blockIdx.x, t = threadIdx.x;
  const int f0 = t * 4;
  const float* pn = nbr  + (size_t)n * HD + f0;
  const float* ph = hbuf + (size_t)n * HD + f0;
  float s = 0.0f;
#pragma unroll
  for (int j = 0; j < 4; ++j)
    s += pn[j] * w_rel[f0 + j] + ph[j] * w_root[f0 + j];
  red[t] = s;
  __syncthreads();
  for (int off = 128; off > 0; off >>= 1) {
    if (t < off) red[t] += red[t + off];
    __syncthreads();
  }
  if (t == 0) score[n] = tanhf(red[0] + b_sag[0]);
}

// ---------------- exact sorted top-k via rank counting (matches lax.top_k) -----
__global__ void __launch_bounds__(256)
topk_k(const float* __restrict__ score, int* __restrict__ oidx,
       float* __restrict__ oval) {
  __shared__ float sm[4096];
  const int n = blockIdx.x * 256 + threadIdx.x;
  const float s = score[n];
  int rank = 0;
  for (int base = 0; base < N_NODES; base += 4096) {
    __syncthreads();
    for (int j = threadIdx.x; j < 4096; j += 256) sm[j] = score[base + j];
    __syncthreads();
    for (int j = 0; j < 4096; ++j) {
      float sj = sm[j];
      int g = base + j;
      rank += (sj > s) || (sj == s && g < n);     // stable tie-break: lower idx first
    }
  }
  if (rank < K_POOL) { oidx[rank] = n; oval[rank] = s; }
}

// ---------------- out[r] = relu(h[idx]*val) @ w_out + b_out ----------------
__global__ void __launch_bounds__(256)
outhead_k(const float* __restrict__ hbuf, const int* __restrict__ oidx,
          const float* __restrict__ oval, const float* __restrict__ w_out,
          const float* __restrict__ b_out, float* __restrict__ out) {
  __shared__ float red[256];
  const int r = blockIdx.x, t = threadIdx.x;
  const int node = oidx[r];
  const float val = oval[r];
  const int f0 = t * 4;
  const float* ph = hbuf + (size_t)node * HD + f0;
  float s = 0.0f;
#pragma unroll
  for (int j = 0; j < 4; ++j) {
    float xp = ph[j] * val;
    xp = (xp > 0.0f) ? xp : 0.0f;
    s += xp * w_out[f0 + j];
  }
  red[t] = s;
  __syncthreads();
  for (int off = 128; off > 0; off >>= 1) {
    if (t < off) red[t] += red[t + off];
    __syncthreads();
  }
  if (t == 0) out[r] = red[0] + b_out[0];
}

extern "C" void kernel_launch(void* const* d_in, const int* in_sizes, int n_in,
                              void* d_out, int out_size, void* d_ws, size_t ws_size,
                              hipStream_t stream) {
  const float* x      = (const float*)d_in[0];
  const int*   ei     = (const int*)d_in[1];
  // d_in[2] edge_attr: unused by reference math
  const float* Wl     = (const float*)d_in[3];
  const float* Wr     = (const float*)d_in[4];
  const float* att    = (const float*)d_in[5];
  const float* b1     = (const float*)d_in[6];
  const float* tptr   = (const float*)d_in[7];
  const float* w_rel  = (const float*)d_in[8];
  const float* w_root = (const float*)d_in[9];
  const float* b_sag  = (const float*)d_in[10];
  const float* w_out  = (const float*)d_in[11];
  const float* b_out  = (const float*)d_in[12];
  float* out = (float*)d_out;

  // ---- workspace layout (floats) ----
  float* ws = (float*)d_ws;
  size_t o = 0;
  float* xl    = ws + o; o += (size_t)N_NODES * HD;     // 64 MB
  float* xr    = ws + o; o += (size_t)N_NODES * HD;     // 64 MB
  float* smax  = ws + o; o += (size_t)N_NODES * HD;     // 64 MB  (reused as h)
  float* numer = ws + o; o += (size_t)N_NODES * HD;     // 64 MB  (reused as nbr)
  float* wsum  = ws + o; o += (size_t)N_NODES * HD;     // 64 MB
  float* logits= ws + o; o += (size_t)N_EDGES * HEADS;  // 8 MB   (reused as ex)
  float* lmax  = ws + o; o += (size_t)N_NODES * HEADS;  // 1 MB
  float* denom = ws + o; o += (size_t)N_NODES * HEADS;  // 1 MB
  float* score = ws + o; o += N_NODES;
  float* oval  = ws + o; o += K_POOL;
  int*   oidx  = (int*)(ws + o); o += K_POOL;
  float* hbuf = smax;    // aliases after their producer/consumer ordering
  float* nbr  = numer;

  const dim3 blk(256);

  init_k<<<dim3((N_NODES * HD) / 256), blk, 0, stream>>>(lmax, denom, smax, wsum, numer);

  gemm_xw_k<<<dim3(1024 * 8, 2), blk, 0, stream>>>(x, Wl, Wr, xl, xr);

  edge_logits_k<<<dim3(N_EDGES), blk, 0, stream>>>(xl, xr, ei, att, logits);

  edge_lmax_k<<<dim3((N_EDGES * HEADS) / 256), blk, 0, stream>>>(logits, ei, lmax);

  edge_exp_k<<<dim3((N_EDGES * HEADS) / 256), blk, 0, stream>>>(logits, ei, lmax, denom);

  pass_smax_k<<<dim3(N_EDGES), blk, 0, stream>>>(xl, ei, logits, denom, tptr, smax);

  pass_accum_k<<<dim3(N_EDGES), blk, 0, stream>>>(xl, ei, logits, denom, tptr, smax,
                                                  wsum, numer);

  node_h_k<<<dim3((N_NODES * HD) / 256), blk, 0, stream>>>(numer, wsum, b1, hbuf);

  pass_nbr_k<<<dim3(N_EDGES), blk, 0, stream>>>(hbuf, ei, nbr);

  node_score_k<<<dim3(N_NODES), blk, 0, stream>>>(nbr, hbuf, w_rel, w_root, b_sag, score);

  topk_k<<<dim3(N_NODES / 256), blk, 0, stream>>>(score, oidx, oval);

  outhead_k<<<dim3(K_POOL), blk, 0, stream>>>(hbuf, oidx, oval, w_out, b_out, out);
}